// FasterRCNN_74603581931869
// MI455X (gfx1250) — compile-verified
//
#include <hip/hip_runtime.h>
#include <stdint.h>

#define NBOX     4096
#define BATCH    8
#define THREADS  1024
#define CHUNK    512
#define CWORDS   (CHUNK / 32)    // 16 words per mask row
#define NCHUNK   (NBOX / CHUNK)  // 8 chunks
#define SCORE_THR 0.5f
#define IOU_THR   0.5f

// One LDS block so asm LDS byte offsets are compile-time constants
// (static __shared__ allocation begins at LDS offset 0).
struct SMem {
  unsigned long long key [NBOX];           // (score_bits<<32)|~idx : 32 KB
  float4             box [NBOX];           // x1,y1,x2,y2 sorted    : 64 KB
  unsigned           keep[NBOX];           //                        16 KB
  unsigned           cmask[CHUNK][CWORDS]; // intra-chunk sup bits   32 KB
};                                         // total 144 KB (< 320 KB/WGP)

__device__ __forceinline__ bool sup_pair(const float4 a, const float4 b) {
  float iw = fminf(a.z, b.z) - fmaxf(a.x, b.x);
  float ih = fminf(a.w, b.w) - fmaxf(a.y, b.y);
  iw = fmaxf(iw, 0.0f);
  ih = fmaxf(ih, 0.0f);
  float inter = iw * ih;
  float uni = (a.z - a.x) * (a.w - a.y) + (b.z - b.x) * (b.w - b.y) - inter;
  return inter > IOU_THR * uni;   // == IoU > thr since union > 0 (wh in [8,96])
}

__global__ __launch_bounds__(THREADS)
void FasterRCNN_nms_kernel(const float* __restrict__ preds,
                           float* __restrict__ out) {
  __shared__ SMem sm;
  const int b   = blockIdx.x;
  const int tid = threadIdx.x;
  const float* batch  = preds + (size_t)b * NBOX * 6;
  float*       obatch = out   + (size_t)b * NBOX * 6;

  // ---- 1) sort keys: descending score, ties -> smaller original idx (stable)
  for (int t = tid; t < NBOX; t += THREADS) {
    float s = batch[t * 6 + 0];  // uniform[0,1): bit-order == value-order
    sm.key[t] = ((unsigned long long)__float_as_uint(s) << 32) | (unsigned)(~t);
  }

  // ---- 2) bitonic sort (descending) fully in LDS: 78 barrier passes
  for (int k = 2; k <= NBOX; k <<= 1) {
    for (int st = k >> 1; st > 0; st >>= 1) {
      __syncthreads();
      for (int t = tid; t < NBOX; t += THREADS) {
        int p = t ^ st;
        if (p > t) {
          unsigned long long a = sm.key[t];
          unsigned long long c = sm.key[p];
          bool desc = ((t & k) == 0);
          if (desc ? (a < c) : (a > c)) { sm.key[t] = c; sm.key[p] = a; }
        }
      }
    }
  }
  __syncthreads();

  // ---- 3) sorted-order gather of [x1,y1,x2,y2] straight into LDS via the
  //         CDNA5 async-to-LDS path. Rows are 24B ([sc,cls,x1,y1,x2,y2]);
  //         +8B -> 4 contiguous box floats, 8B aligned -> 2x B64.
  //         INST_OFFSET is added to BOTH the global and the LDS address.
  {
    constexpr unsigned BOXOFF = (unsigned)__builtin_offsetof(SMem, box);
    for (int t = tid; t < NBOX; t += THREADS) {
      unsigned orig = ~(unsigned)(sm.key[t]);           // low 32 bits = ~idx
      const float* src = batch + (size_t)orig * 6 + 2;  // &x1
      unsigned long long ga = (unsigned long long)(uintptr_t)src;
      unsigned ldsa = BOXOFF + 16u * (unsigned)t;
      asm volatile("global_load_async_to_lds_b64 %0, %1, off"
                   :: "v"(ldsa), "v"(ga) : "memory");
      asm volatile("global_load_async_to_lds_b64 %0, %1, off offset:8"
                   :: "v"(ldsa), "v"(ga) : "memory");
    }
    asm volatile("s_wait_asynccnt 0x0" ::: "memory");
  }
  __syncthreads();

  // ---- 4) initial keep = sorted score > SCORE_THR
  for (int t = tid; t < NBOX; t += THREADS) {
    float sc = __uint_as_float((unsigned)(sm.key[t] >> 32));
    sm.keep[t] = (sc > SCORE_THR) ? 1u : 0u;
  }

  // ---- 5) exact greedy NMS, chunked: 3 barriers/chunk instead of 1/box.
  for (int c = 0; c < NCHUNK; ++c) {
    const int cb = c * CHUNK;
    __syncthreads();

    // (a) parallel: intra-chunk suppression bitmatrix (bit = j>i && IoU>thr)
    for (int t = tid; t < CHUNK * CWORDS; t += THREADS) {
      int ii = t >> 4;            // chunk-local row   (t / CWORDS)
      int w  = t & (CWORDS - 1);  // word within row
      float4 bi = sm.box[cb + ii];
      unsigned bits = 0u;
      int jbase = w * 32;
      for (int bt = 0; bt < 32; ++bt) {
        int jj = jbase + bt;
        if (jj > ii && sup_pair(bi, sm.box[cb + jj])) bits |= (1u << bt);
      }
      sm.cmask[ii][w] = bits;
    }
    __syncthreads();

    // (b) serial greedy recurrence inside the chunk, ONE wave32, lockstep:
    //     lane owns a 32-bit column word; per step: shfl-broadcast alive
    //     word from owner lane, then OR the suppressor's mask row.
    if (tid < 32) {
      const int lane = tid;
      const int myw = lane & (CWORDS - 1);  // lanes 16..31 mirror 0..15
      unsigned kw = 0u;                     // alive-at-chunk-entry bits
      for (int bt = 0; bt < 32; ++bt)
        kw |= (sm.keep[cb + myw * 32 + bt] ? 1u : 0u) << bt;
      unsigned rem = 0u;                    // suppressed-within-chunk bits
      for (int ii = 0; ii < CHUNK; ++ii) {
        unsigned aliveword = __shfl(kw & ~rem, ii >> 5, 32);
        if ((aliveword >> (ii & 31)) & 1u)  // uniform branch (broadcast)
          rem |= sm.cmask[ii][myw];
      }
      if (lane < CWORDS) {
        for (int bt = 0; bt < 32; ++bt)
          if ((rem >> bt) & 1u) sm.keep[cb + myw * 32 + bt] = 0u;
      }
    }
    __syncthreads();

    // (c) parallel: chunk survivors suppress all later boxes
    for (int j = cb + CHUNK + tid; j < NBOX; j += THREADS) {
      if (!sm.keep[j]) continue;
      float4 bj = sm.box[j];
      for (int ii = 0; ii < CHUNK; ++ii) {
        if (sm.keep[cb + ii] && sup_pair(sm.box[cb + ii], bj)) {
          sm.keep[j] = 0u;
          break;
        }
      }
    }
  }
  __syncthreads();

  // ---- 6) emit sorted rows * keep (score/cls re-gathered through L2)
  for (int t = tid; t < NBOX; t += THREADS) {
    unsigned orig = ~(unsigned)(sm.key[t]);
    float m = sm.keep[t] ? 1.0f : 0.0f;
    const float* src = batch + (size_t)orig * 6;
    float4 bx = sm.box[t];
    float* dst = obatch + (size_t)t * 6;
    dst[0] = src[0] * m;   // score
    dst[1] = src[1] * m;   // cls
    dst[2] = bx.x * m;
    dst[3] = bx.y * m;
    dst[4] = bx.z * m;
    dst[5] = bx.w * m;
  }
}

extern "C" void kernel_launch(void* const* d_in, const int* in_sizes, int n_in,
                              void* d_out, int out_size, void* d_ws, size_t ws_size,
                              hipStream_t stream) {
  (void)in_sizes; (void)n_in; (void)d_ws; (void)ws_size; (void)out_size;
  const float* preds = (const float*)d_in[0];   // (8, 4096, 6) float32
  float*       out   = (float*)d_out;           // (8, 4096, 6) float32
  FasterRCNN_nms_kernel<<<dim3(BATCH), dim3(THREADS), 0, stream>>>(preds, out);
}